// TestGNN_87978110091596
// MI455X (gfx1250) — compile-verified
//
#include <hip/hip_runtime.h>

#define NN 100000
#define EE 600000
#define GG 1024
#define AA 74
#define BB 12
#define DD 128
#define DEPTH 3

typedef __attribute__((ext_vector_type(2))) float v2f;
typedef __attribute__((ext_vector_type(8))) float v8f;

__device__ inline void atomicAddF(float* p, float v) {
    // lowers to global_atomic_add_f32 (no return) on CDNA5
    __hip_atomic_fetch_add(p, v, __ATOMIC_RELAXED, __HIP_MEMORY_SCOPE_AGENT);
}

// ---------------------------------------------------------------- zero fill
__global__ void fill_zero(float* __restrict__ p, int n) {
    int i = blockIdx.x * blockDim.x + threadIdx.x;
    const int stride = gridDim.x * blockDim.x;
    for (; i < n; i += stride) p[i] = 0.0f;
}

// ---------------------------------------------------------------- atom embedding
// x = atomic_num[N,74] @ W_atom[74,128], K padded to 76 with zeros in A-tile.
#define EMB_MT 5
#define EMB_KH 38   // ceil(74/2) padded to even K=76 -> 38 float2 per row
#define EMB_ZS 40   // padded LDS stride (float2 units)

__global__ void __launch_bounds__(256)
embed_kernel(const float* __restrict__ atom, const float* __restrict__ Wa,
             float* __restrict__ xout) {
    __shared__ float2 Zl[16 * EMB_MT * EMB_ZS];
    const int tid = threadIdx.x;
    const int m0  = blockIdx.x * (16 * EMB_MT);

    for (int idx = tid; idx < 16 * EMB_MT * EMB_KH; idx += 256) {
        const int r = idx / EMB_KH, c = idx % EMB_KH;
        const int node = m0 + r;
        const int k0 = 2 * c;
        const float z0 = (k0     < AA) ? atom[node * AA + k0]     : 0.0f;
        const float z1 = (k0 + 1 < AA) ? atom[node * AA + k0 + 1] : 0.0f;
        Zl[r * EMB_ZS + c] = make_float2(z0, z1);
    }
    __syncthreads();

    const int lane = tid & 31, wave = tid >> 5;
    const int lo = lane & 15, hi = lane >> 4;
    const int col = wave * 16 + lo;

    for (int mt = 0; mt < EMB_MT; ++mt) {
        v8f acc = {};
        const float2* zrow = &Zl[(mt * 16 + lo) * EMB_ZS];
#pragma unroll
        for (int kh = 0; kh < EMB_KH; kh += 2) {   // K step 4
            const float2 az = zrow[kh + hi];
            v2f a; a[0] = az.x; a[1] = az.y;
            const int k0  = 2 * (kh + hi);
            const int k0c = (k0     < AA) ? k0       : (AA - 1); // A is zero there
            const int k1c = (k0 + 1 < AA) ? (k0 + 1) : (AA - 1);
            v2f b; b[0] = Wa[k0c * DD + col]; b[1] = Wa[k1c * DD + col];
            acc = __builtin_amdgcn_wmma_f32_16x16x4_f32(
                false, a, false, b, (short)0, acc, false, false);
        }
#pragma unroll
        for (int r = 0; r < 8; ++r) {
            const int row = m0 + mt * 16 + r + (hi ? 8 : 0);
            xout[row * DD + col] = acc[r];
        }
    }
}

// ---------------------------------------------------------------- edge pass
// msg = relu(x[src] + bond@W_bond), atomic sum-aggregate into agg[dst].
#define EPB 64
__global__ void __launch_bounds__(128)
edge_kernel(const float* __restrict__ x, const float* __restrict__ bond,
            const float* __restrict__ Wb, const int* __restrict__ src,
            const int* __restrict__ dst, float* __restrict__ agg) {
    __shared__ float Wbs[BB * DD];
    __shared__ float bls[EPB * BB];
    __shared__ int   ss[EPB], ds[EPB];
    const int t  = threadIdx.x;           // feature dim 0..127
    const int e0 = blockIdx.x * EPB;

    for (int i = t; i < BB * DD; i += 128)  Wbs[i] = Wb[i];
    for (int i = t; i < EPB * BB; i += 128) bls[i] = bond[e0 * BB + i];
    if (t < EPB) { ss[t] = src[e0 + t]; ds[t] = dst[e0 + t]; }
    __syncthreads();

    for (int i = 0; i < EPB; ++i) {
        const int s = ss[i], d = ds[i];
        float acc = x[s * DD + t];                 // coalesced 512B gather
        const float* bl = &bls[i * BB];
#pragma unroll
        for (int j = 0; j < BB; ++j) acc += bl[j] * Wbs[j * DD + t];
        acc = fmaxf(acc, 0.0f);
        atomicAddF(&agg[d * DD + t], acc);         // coalesced atomic scatter
    }
}

// ---------------------------------------------------------------- dense layer
// x_out = relu(((1+eps)*x + agg) @ W + b) + x      (residual == layer input)
#define DN_MT 5
#define DN_KH 64    // 128/2 float2 per row
#define DN_ZS 66    // padded LDS stride

__global__ void __launch_bounds__(256)
dense_kernel(const float* __restrict__ x, const float* __restrict__ agg,
             const float* __restrict__ Wall, const float* __restrict__ ball,
             const float* __restrict__ eps, int layer,
             float* __restrict__ xout) {
    __shared__ float2 Zl[16 * DN_MT * DN_ZS];
    const int tid = threadIdx.x;
    const int m0  = blockIdx.x * (16 * DN_MT);
    const float ep1 = 1.0f + eps[layer];
    const float* W  = Wall + (size_t)layer * DD * DD;

    for (int idx = tid; idx < 16 * DN_MT * DN_KH; idx += 256) {
        const int r = idx >> 6, c = idx & 63;
        const int node = m0 + r;
        const float2 xv = *(const float2*)(x   + node * DD + 2 * c);
        const float2 av = *(const float2*)(agg + node * DD + 2 * c);
        Zl[r * DN_ZS + c] = make_float2(ep1 * xv.x + av.x, ep1 * xv.y + av.y);
    }
    __syncthreads();

    const int lane = tid & 31, wave = tid >> 5;
    const int lo = lane & 15, hi = lane >> 4;
    const int col = wave * 16 + lo;
    const float bias = ball[layer * DD + col];

    for (int mt = 0; mt < DN_MT; ++mt) {
        v8f acc = {};
        const float2* zrow = &Zl[(mt * 16 + lo) * DN_ZS];
#pragma unroll
        for (int kh = 0; kh < DN_KH; kh += 2) {    // K step 4, 32 WMMAs
            const float2 az = zrow[kh + hi];
            v2f a; a[0] = az.x; a[1] = az.y;
            const int k0 = 2 * (kh + hi);
            v2f b; b[0] = W[k0 * DD + col]; b[1] = W[(k0 + 1) * DD + col];
            acc = __builtin_amdgcn_wmma_f32_16x16x4_f32(
                false, a, false, b, (short)0, acc, false, false);
        }
#pragma unroll
        for (int r = 0; r < 8; ++r) {
            const int row = m0 + mt * 16 + r + (hi ? 8 : 0);
            const float v = fmaxf(acc[r] + bias, 0.0f) + x[row * DD + col];
            xout[row * DD + col] = v;
        }
    }
}

// ---------------------------------------------------------------- graph pooling
__global__ void __launch_bounds__(128)
pool_scatter(const float* __restrict__ x, const int* __restrict__ n2g,
             float* __restrict__ fsum, float* __restrict__ fcnt) {
    const int t  = threadIdx.x;
    const int n0 = blockIdx.x * 8;
    for (int i = 0; i < 8; ++i) {
        const int n = n0 + i;
        const int g = n2g[n];
        atomicAddF(&fsum[g * DD + t], x[n * DD + t]);
        if (t == 0) atomicAddF(&fcnt[g], 1.0f);
    }
}

__global__ void __launch_bounds__(256)
readout_kernel(const float* __restrict__ fsum, const float* __restrict__ fcnt,
               const float* __restrict__ Wm, const float* __restrict__ bm,
               float* __restrict__ out) {
    const int lane = threadIdx.x & 31;
    const int g = blockIdx.x * 8 + (threadIdx.x >> 5);
    float p = 0.0f;
#pragma unroll
    for (int j = 0; j < 4; ++j) {
        const int d = lane + 32 * j;
        p += fsum[g * DD + d] * Wm[d];
    }
#pragma unroll
    for (int off = 16; off > 0; off >>= 1) p += __shfl_xor(p, off, 32);
    if (lane == 0) out[g] = p / fmaxf(fcnt[g], 1.0f) + bm[0];
}

// ---------------------------------------------------------------- launcher
extern "C" void kernel_launch(void* const* d_in, const int* in_sizes, int n_in,
                              void* d_out, int out_size, void* d_ws, size_t ws_size,
                              hipStream_t stream) {
    (void)in_sizes; (void)n_in; (void)out_size; (void)ws_size;
    const float* atom = (const float*)d_in[0];
    const float* bond = (const float*)d_in[1];
    const float* Wa   = (const float*)d_in[2];
    const float* Wb   = (const float*)d_in[3];
    const float* Wlyr = (const float*)d_in[4];
    const float* blyr = (const float*)d_in[5];
    const float* eps  = (const float*)d_in[6];
    const float* Wm   = (const float*)d_in[7];
    const float* bm   = (const float*)d_in[8];
    const int*   src  = (const int*)d_in[9];
    const int*   dst  = (const int*)d_in[10];
    const int*   n2g  = (const int*)d_in[11];
    float* out = (float*)d_out;

    float* xa   = (float*)d_ws;                 // [N,D]
    float* xb   = xa  + (size_t)NN * DD;        // [N,D]
    float* agg  = xb  + (size_t)NN * DD;        // [N,D]
    float* fsum = agg + (size_t)NN * DD;        // [G,D]
    float* fcnt = fsum + (size_t)GG * DD;       // [G]   (contiguous with fsum)

    embed_kernel<<<NN / (16 * EMB_MT), 256, 0, stream>>>(atom, Wa, xa);

    float* xc = xa; float* xn = xb;
    for (int l = 0; l < DEPTH; ++l) {
        fill_zero<<<2048, 256, 0, stream>>>(agg, NN * DD);
        edge_kernel<<<EE / EPB, 128, 0, stream>>>(xc, bond, Wb, src, dst, agg);
        dense_kernel<<<NN / (16 * DN_MT), 256, 0, stream>>>(xc, agg, Wlyr, blyr, eps, l, xn);
        float* t = xc; xc = xn; xn = t;
    }

    fill_zero<<<64, 256, 0, stream>>>(fsum, GG * DD + GG);
    pool_scatter<<<NN / 8, 128, 0, stream>>>(xc, n2g, fsum, fcnt);
    readout_kernel<<<GG / 8, 256, 0, stream>>>(fsum, fcnt, Wm, bm, out);
}